// MemPool_20040317403727
// MI455X (gfx1250) — compile-verified
//
#include <hip/hip_runtime.h>
#include <hip/hip_bf16.h>

// MI455X / gfx1250, wave32. All GEMMs via V_WMMA_F32_16X16X4_F32 (full fp32
// precision; problem is bandwidth-bound at 23.3 TB/s so fp32 WMMA is free).

typedef float v2f __attribute__((ext_vector_type(2)));
typedef float v8f __attribute__((ext_vector_type(8)));

#define BB   8
#define NN   4096
#define DD   256
#define HH   4
#define KC   64
#define HK   256     // H*K
#define OUTD 256

#define DCHUNK 64    // d-chunk staged in LDS (k_clusters)
#define KST    66    // klds/qlds row stride (floats): 8B-aligned rows, bank-spread
#define DT     260   // dist tile row stride (floats)

#define NSPLIT 8     // n-contraction splits in k_pool
#define NCHUNK 64    // n-chunk staged in LDS (k_pool)
#define CS     68    // C-chunk row stride (floats): 16B-aligned rows, bank-spread

// ---------------------------------------------------------------- row norms
__global__ void k_ksq(const float* __restrict__ kmem, float* __restrict__ ksq) {
    int t = threadIdx.x;                       // 256 threads, one hk row each
    const float4* p = (const float4*)(kmem + (size_t)t * DD);
    float s = 0.f;
    #pragma unroll 8
    for (int j = 0; j < DD / 4; ++j) {
        float4 v = p[j];
        s += v.x * v.x + v.y * v.y + v.z * v.z + v.w * v.w;
    }
    ksq[t] = s;
}

__global__ void k_qsq(const float* __restrict__ q, float* __restrict__ qsq) {
    size_t bn = (size_t)blockIdx.x * blockDim.x + threadIdx.x;  // one (b,n) row
    const float4* p = (const float4*)(q + bn * DD);
    float s = 0.f;
    #pragma unroll 8
    for (int j = 0; j < DD / 4; ++j) {
        float4 v = p[j];
        s += v.x * v.x + v.y * v.y + v.z * v.z + v.w * v.w;
    }
    qsq[bn] = s;
}

// ------------------------------------------------- qk GEMM + full C epilogue
// grid (N/16, B), block 256 (8 waves). Each wave owns two 16x16 hk tiles.
__global__ void k_clusters(const float* __restrict__ q,
                           const int*   __restrict__ mask,
                           const float* __restrict__ kmem,
                           const float* __restrict__ conv,
                           const float* __restrict__ qsq,
                           const float* __restrict__ ksq,
                           float* __restrict__ Cout) {
    const int n0   = blockIdx.x * 16;
    const int b    = blockIdx.y;
    const int t    = threadIdx.x;
    const int w    = t >> 5;
    const int lane = t & 31;
    const int lh   = lane >> 4;     // which half of the wave
    const int lm   = lane & 15;

    extern __shared__ float smem[];
    float* klds  = smem;                   // [HK][KST]
    float* qlds  = klds + HK * KST;        // [16][KST]
    float* dtile = qlds + 16 * KST;        // [16][DT]
    float* hsum  = dtile + 16 * DT;        // [16][H] reciprocal head sums

    const int hk0 = (w * 2) * 16;
    const int hk1 = hk0 + 16;

    v8f acc0 = {};
    v8f acc1 = {};

    for (int dc = 0; dc < DD; dc += DCHUNK) {
        // stage kmem[:, dc:dc+64]: thread t loads its own hk row
        {
            const float4* src = (const float4*)(kmem + (size_t)t * DD + dc);
            float* dst = klds + t * KST;
            #pragma unroll
            for (int j = 0; j < DCHUNK / 4; ++j) {
                float4 v = src[j];
                dst[j * 4 + 0] = v.x; dst[j * 4 + 1] = v.y;
                dst[j * 4 + 2] = v.z; dst[j * 4 + 3] = v.w;
            }
        }
        // stage q[b, n0:n0+16, dc:dc+64]
        {
            int idx = t * 4;              // 1024 floats total
            int row = idx >> 6;
            int col = idx & 63;
            float4 v = *(const float4*)(q + ((size_t)b * NN + n0 + row) * DD + dc + col);
            float* dst = qlds + row * KST + col;
            dst[0] = v.x; dst[1] = v.y; dst[2] = v.z; dst[3] = v.w;
        }
        if (dc + DCHUNK < DD)             // warm next kmem chunk
            __builtin_prefetch(kmem + (size_t)t * DD + dc + DCHUNK, 0, 3);
        __syncthreads();

        #pragma unroll 4
        for (int dk = 0; dk < DCHUNK; dk += 4) {
            const int ko = dk + (lh << 1);
            v2f a;  a.x  = qlds[lm * KST + ko];          a.y  = qlds[lm * KST + ko + 1];
            v2f b0; b0.x = klds[(hk0 + lm) * KST + ko];  b0.y = klds[(hk0 + lm) * KST + ko + 1];
            v2f b1; b1.x = klds[(hk1 + lm) * KST + ko];  b1.y = klds[(hk1 + lm) * KST + ko + 1];
            acc0 = __builtin_amdgcn_wmma_f32_16x16x4_f32(false, a, false, b0, (short)0, acc0, false, false);
            acc1 = __builtin_amdgcn_wmma_f32_16x16x4_f32(false, a, false, b1, (short)0, acc1, false, false);
        }
        __syncthreads();
    }

    // dist = 1/(1 + d2)  (TAU == 1  =>  power -(TAU+1)/2 == -1)
    float qs[8];
    #pragma unroll
    for (int r = 0; r < 8; ++r)
        qs[r] = qsq[(size_t)b * NN + n0 + r + (lh ? 8 : 0)];
    const float ks0 = ksq[hk0 + lm];
    const float ks1 = ksq[hk1 + lm];
    #pragma unroll
    for (int r = 0; r < 8; ++r) {
        int M = r + (lh ? 8 : 0);
        float d2a = fmaxf(qs[r] + ks0 - 2.f * acc0[r], 0.f);
        float d2b = fmaxf(qs[r] + ks1 - 2.f * acc1[r], 0.f);
        dtile[M * DT + hk0 + lm] = 1.f / (1.f + d2a);
        dtile[M * DT + hk1 + lm] = 1.f / (1.f + d2b);
    }
    __syncthreads();

    // reciprocal per-head sums over K
    if (t < 16 * HH) {
        int n = t >> 2, h = t & 3;
        float s = 0.f;
        #pragma unroll 8
        for (int k = 0; k < KC; ++k) s += dtile[n * DT + h * KC + k];
        hsum[n * HH + h] = 1.f / s;
    }
    __syncthreads();

    // head-mix + softmax over K + mask; lanes 0-15 / 16-31 each own one n row
    {
        int n  = t >> 4;        // 0..15
        int kg = t & 15;        // 4 k's per thread
        float c0 = conv[0], c1 = conv[1], c2 = conv[2], c3 = conv[3];
        float i0 = hsum[n * HH + 0], i1 = hsum[n * HH + 1];
        float i2 = hsum[n * HH + 2], i3 = hsum[n * HH + 3];
        float mixed[4];
        #pragma unroll
        for (int j = 0; j < 4; ++j) {
            int k = kg * 4 + j;
            mixed[j] = c0 * dtile[n * DT + 0 * KC + k] * i0
                     + c1 * dtile[n * DT + 1 * KC + k] * i1
                     + c2 * dtile[n * DT + 2 * KC + k] * i2
                     + c3 * dtile[n * DT + 3 * KC + k] * i3;
        }
        float vmax = fmaxf(fmaxf(mixed[0], mixed[1]), fmaxf(mixed[2], mixed[3]));
        #pragma unroll
        for (int m = 1; m <= 8; m <<= 1) vmax = fmaxf(vmax, __shfl_xor(vmax, m, 32));
        float e[4], es = 0.f;
        #pragma unroll
        for (int j = 0; j < 4; ++j) { e[j] = __expf(mixed[j] - vmax); es += e[j]; }
        #pragma unroll
        for (int m = 1; m <= 8; m <<= 1) es += __shfl_xor(es, m, 32);
        float mk  = (float)mask[(size_t)b * NN + n0 + n];
        float inv = mk / es;
        #pragma unroll
        for (int j = 0; j < 4; ++j)
            Cout[((size_t)b * KC + kg * 4 + j) * NN + n0 + n] = e[j] * inv;
    }
}

// --------------------------------------- q_l[b,k,d] = sum_n C[b,k,n] q[b,n,d]
// grid (NSPLIT, B), block 256 (8 waves). Each wave: one 16-row k-tile, eight
// 16-col d-tiles -> 8 independent WMMA accumulator chains, A fragment reused 8x.
// LDS stages q[nc:nc+64, :] and C[:, nc:nc+64] per chunk. Partial sums over the
// wave's n-range go to ws; k_reduce folds the NSPLIT partials (deterministic).
__global__ void k_pool(const float* __restrict__ Cmat,
                       const float* __restrict__ q,
                       float* __restrict__ part) {
    const int b    = blockIdx.y;
    const int s    = blockIdx.x;           // n-split
    const int t    = threadIdx.x;
    const int w    = t >> 5;
    const int lane = t & 31;
    const int lh   = lane >> 4, lm = lane & 15;
    const int mt   = w & 3;                // k-tile of this wave
    const int ntb  = (w >> 2) * 8;         // first of 8 d-tiles

    extern __shared__ float smem[];
    float* qlds = smem;                    // [NCHUNK][DD]
    float* clds = qlds + NCHUNK * DD;      // [KC][CS]

    v8f acc[8];
    #pragma unroll
    for (int i = 0; i < 8; ++i) acc[i] = (v8f){};

    const int nc0 = s * (NN / NSPLIT);
    for (int nc = nc0; nc < nc0 + NN / NSPLIT; nc += NCHUNK) {
        {   // stage q[b, nc:nc+64, :]  (64x256 floats)
            int row = t >> 2, colb = (t & 3) * 64;
            const float* src = q + ((size_t)b * NN + nc + row) * DD + colb;
            float* dst = qlds + row * DD + colb;
            #pragma unroll
            for (int j = 0; j < 16; ++j)
                *(float4*)(dst + j * 4) = *(const float4*)(src + j * 4);
        }
        {   // stage C[b, :, nc:nc+64]  (64x64 floats, padded rows)
            int row = t >> 2, colb = (t & 3) * 16;
            const float* src = Cmat + ((size_t)b * KC + row) * NN + nc + colb;
            float* dst = clds + row * CS + colb;
            #pragma unroll
            for (int j = 0; j < 4; ++j)
                *(float4*)(dst + j * 4) = *(const float4*)(src + j * 4);
        }
        __syncthreads();

        for (int dk = 0; dk < NCHUNK; dk += 4) {
            const int ko = dk + (lh << 1);
            v2f a; a.x = clds[(mt * 16 + lm) * CS + ko];
                   a.y = clds[(mt * 16 + lm) * CS + ko + 1];
            #pragma unroll
            for (int i = 0; i < 8; ++i) {
                const int d0 = (ntb + i) * 16;
                v2f bb; bb.x = qlds[ko * DD + d0 + lm];
                        bb.y = qlds[(ko + 1) * DD + d0 + lm];
                acc[i] = __builtin_amdgcn_wmma_f32_16x16x4_f32(false, a, false, bb, (short)0, acc[i], false, false);
            }
        }
        __syncthreads();
    }

    float* dst = part + (size_t)s * BB * KC * DD;
    #pragma unroll
    for (int i = 0; i < 8; ++i) {
        #pragma unroll
        for (int r = 0; r < 8; ++r) {
            int m = r + (lh ? 8 : 0);
            dst[((size_t)b * KC + mt * 16 + m) * DD + (ntb + i) * 16 + lm] = acc[i][r];
        }
    }
}

// fold the NSPLIT partial q_l buffers (deterministic fixed-order sum)
__global__ void k_reduce(const float* __restrict__ part, float* __restrict__ ql) {
    size_t i = (size_t)blockIdx.x * blockDim.x + threadIdx.x;
    float s = 0.f;
    #pragma unroll
    for (int j = 0; j < NSPLIT; ++j) s += part[(size_t)j * BB * KC * DD + i];
    ql[i] = s;
}

// ------------------------------------------- v = leaky_relu(q_l @ w)
// grid (4, B), block 256: wave owns one k-tile + two out-tiles (A reuse, 2 chains)
__global__ void k_out(const float* __restrict__ ql,
                      const float* __restrict__ w,
                      float* __restrict__ vout) {
    const int b    = blockIdx.y;
    const int wv   = threadIdx.x >> 5;
    const int lane = threadIdx.x & 31;
    const int lh   = lane >> 4, lm = lane & 15;
    const int mt   = wv >> 1;                        // k-tile 0..3
    const int nt0  = blockIdx.x * 4 + (wv & 1) * 2;  // out-tile pair
    const int nt1  = nt0 + 1;

    const float* arow = ql + ((size_t)b * KC + mt * 16 + lm) * DD;
    const float* wb0  = w + nt0 * 16 + lm;
    const float* wb1  = w + nt1 * 16 + lm;
    v8f acc0 = {}, acc1 = {};
    #pragma unroll 4
    for (int kk = 0; kk < DD; kk += 4) {
        int ko = kk + (lh << 1);
        v2f a;  a.x  = arow[ko];                a.y  = arow[ko + 1];
        v2f b0; b0.x = wb0[(size_t)ko * OUTD];  b0.y = wb0[(size_t)(ko + 1) * OUTD];
        v2f b1; b1.x = wb1[(size_t)ko * OUTD];  b1.y = wb1[(size_t)(ko + 1) * OUTD];
        acc0 = __builtin_amdgcn_wmma_f32_16x16x4_f32(false, a, false, b0, (short)0, acc0, false, false);
        acc1 = __builtin_amdgcn_wmma_f32_16x16x4_f32(false, a, false, b1, (short)0, acc1, false, false);
    }
    #pragma unroll
    for (int r = 0; r < 8; ++r) {
        int m = r + (lh ? 8 : 0);
        float x0 = acc0[r]; x0 = (x0 >= 0.f) ? x0 : 0.01f * x0;
        float x1 = acc1[r]; x1 = (x1 >= 0.f) ? x1 : 0.01f * x1;
        vout[((size_t)b * KC + mt * 16 + m) * OUTD + nt0 * 16 + lm] = x0;
        vout[((size_t)b * KC + mt * 16 + m) * OUTD + nt1 * 16 + lm] = x1;
    }
}

extern "C" void kernel_launch(void* const* d_in, const int* in_sizes, int n_in,
                              void* d_out, int out_size, void* d_ws, size_t ws_size,
                              hipStream_t stream) {
    (void)in_sizes; (void)n_in; (void)out_size; (void)ws_size;
    const float* q    = (const float*)d_in[0];
    const int*   mask = (const int*)d_in[1];
    const float* kmem = (const float*)d_in[2];
    const float* conv = (const float*)d_in[3];
    const float* w    = (const float*)d_in[4];

    float* vout = (float*)d_out;                         // [B,K,OUT] = 131072
    float* Cout = vout + (size_t)BB * KC * OUTD;         // [B,K,N]   = 2097152

    float* ws   = (float*)d_ws;
    float* qsq  = ws;                                    // B*N             = 32768
    float* ksq  = qsq + (size_t)BB * NN;                 // HK              = 256
    float* ql   = ksq + HK;                              // B*K*D           = 131072
    float* part = ql + (size_t)BB * KC * DD;             // NSPLIT*B*K*D    = 1048576

    k_ksq<<<1, 256, 0, stream>>>(kmem, ksq);
    k_qsq<<<BB * NN / 256, 256, 0, stream>>>(q, qsq);

    size_t shm1 = (size_t)(HK * KST + 16 * KST + 16 * DT + 16 * HH) * sizeof(float);
    k_clusters<<<dim3(NN / 16, BB), 256, shm1, stream>>>(q, mask, kmem, conv, qsq, ksq, Cout);

    size_t shm2 = (size_t)(NCHUNK * DD + KC * CS) * sizeof(float);
    k_pool<<<dim3(NSPLIT, BB), 256, shm2, stream>>>(Cout, q, part);
    k_reduce<<<BB * KC * DD / 256, 256, 0, stream>>>(part, ql);

    k_out<<<dim3(4, BB), 256, 0, stream>>>(ql, w, vout);
}